// encoder_33973191311953
// MI455X (gfx1250) — compile-verified
//
#include <hip/hip_runtime.h>
#include <math.h>

// ---------------------------------------------------------------------------
// Problem constants (from reference): B=256, T=60, Din=2048, D=H=512, 4H=2048
// ---------------------------------------------------------------------------
#define B_   256
#define T_   60
#define DIN  2048
#define D_   512
#define H_   512
#define G4   2048   // 4*H

typedef float v2f __attribute__((ext_vector_type(2)));
typedef float v8f __attribute__((ext_vector_type(8)));

// fp32 WMMA: D(16x16) = A(16x4) * B(4x16) + C   -> v_wmma_f32_16x16x4_f32
__device__ __forceinline__ v8f wmma4(v2f a, v2f b, v8f c) {
  return __builtin_amdgcn_wmma_f32_16x16x4_f32(
      /*neg_a=*/false, a, /*neg_b=*/false, b,
      /*c_mod=*/(short)0, c, /*reuse_a=*/false, /*reuse_b=*/false);
}

__device__ __forceinline__ float sigm(float x) { return 1.0f / (1.0f + expf(-x)); }

// ---------------------------------------------------------------------------
// Zero h, c and BN accumulators (graph-replay safe: re-initialized each call)
// ---------------------------------------------------------------------------
__global__ __launch_bounds__(256) void k_zero(float* h, float* c,
                                              float* sum, float* sumsq) {
  int i = blockIdx.x * blockDim.x + threadIdx.x;
  if (i < B_ * H_) { h[i] = 0.0f; c[i] = 0.0f; }
  if (i < D_)      { sum[i] = 0.0f; sumsq[i] = 0.0f; }
}

// ---------------------------------------------------------------------------
// Encoder GEMM: V[m,n] = X[m,:] . Wenc[n,:] + benc[n];  M=15360, N=512, K=2048
// Wave strip: 32 rows x 64 cols (2 M-tiles x 4 N-tiles -> 8 accumulators).
// Each B fragment is reused by two WMMAs: 6 b64 loads per 8 WMMAs.
// Block: 8 waves = 256x64 tile. Grid: (15360/256, 512/64) = (60, 8).
// ---------------------------------------------------------------------------
__global__ __launch_bounds__(256) void k_encode(const float* __restrict__ X,
                                                const float* __restrict__ Wenc,
                                                const float* __restrict__ benc,
                                                float* __restrict__ V) {
  const int lane = threadIdx.x & 31;
  const int wave = threadIdx.x >> 5;
  const int lo = lane & 15;          // m (for A) / n (for B/D) within tile
  const int hi = lane >> 4;          // K-half selector for A/B, M-half for D
  const int mb = blockIdx.x * 256 + wave * 32;
  const int nb = blockIdx.y * 64;

  const float* a_row0 = X    + (size_t)(mb +  0 + lo) * DIN;
  const float* a_row1 = X    + (size_t)(mb + 16 + lo) * DIN;
  const float* b_row0 = Wenc + (size_t)(nb +  0 + lo) * DIN;
  const float* b_row1 = Wenc + (size_t)(nb + 16 + lo) * DIN;
  const float* b_row2 = Wenc + (size_t)(nb + 32 + lo) * DIN;
  const float* b_row3 = Wenc + (size_t)(nb + 48 + lo) * DIN;
  const int koff = 2 * hi;

  v8f p0 = {}, p1 = {}, p2 = {}, p3 = {};   // rows mb..mb+15
  v8f q0 = {}, q1 = {}, q2 = {}, q3 = {};   // rows mb+16..mb+31
  for (int kb = 0; kb < DIN; kb += 4) {
    if ((kb & 63) == 0) {            // stream next 256B/row ahead of the MACs
      __builtin_prefetch(a_row0 + kb + 64, 0, 1);
      __builtin_prefetch(a_row1 + kb + 64, 0, 1);
      __builtin_prefetch(b_row0 + kb + 64, 0, 1);
      __builtin_prefetch(b_row1 + kb + 64, 0, 1);
      __builtin_prefetch(b_row2 + kb + 64, 0, 1);
      __builtin_prefetch(b_row3 + kb + 64, 0, 1);
    }
    v2f a0 = *(const v2f*)(a_row0 + kb + koff);
    v2f a1 = *(const v2f*)(a_row1 + kb + koff);
    v2f b0 = *(const v2f*)(b_row0 + kb + koff);
    v2f b1 = *(const v2f*)(b_row1 + kb + koff);
    v2f b2 = *(const v2f*)(b_row2 + kb + koff);
    v2f b3 = *(const v2f*)(b_row3 + kb + koff);
    p0 = wmma4(a0, b0, p0);  q0 = wmma4(a1, b0, q0);
    p1 = wmma4(a0, b1, p1);  q1 = wmma4(a1, b1, q1);
    p2 = wmma4(a0, b2, p2);  q2 = wmma4(a1, b2, q2);
    p3 = wmma4(a0, b3, p3);  q3 = wmma4(a1, b3, q3);
  }

  const float bias0 = benc[nb +  0 + lo];
  const float bias1 = benc[nb + 16 + lo];
  const float bias2 = benc[nb + 32 + lo];
  const float bias3 = benc[nb + 48 + lo];
#pragma unroll
  for (int r = 0; r < 8; ++r) {
    const size_t m0 = (size_t)(mb + r + 8 * hi);
    const size_t m1 = m0 + 16;
    V[m0 * D_ + nb +  0 + lo] = p0[r] + bias0;
    V[m0 * D_ + nb + 16 + lo] = p1[r] + bias1;
    V[m0 * D_ + nb + 32 + lo] = p2[r] + bias2;
    V[m0 * D_ + nb + 48 + lo] = p3[r] + bias3;
    V[m1 * D_ + nb +  0 + lo] = q0[r] + bias0;
    V[m1 * D_ + nb + 16 + lo] = q1[r] + bias1;
    V[m1 * D_ + nb + 32 + lo] = q2[r] + bias2;
    V[m1 * D_ + nb + 48 + lo] = q3[r] + bias3;
  }
}

// ---------------------------------------------------------------------------
// BatchNorm stats over (B*T) rows per channel d (coalesced; atomic combine)
// ---------------------------------------------------------------------------
__global__ __launch_bounds__(512) void k_stats(const float* __restrict__ V,
                                               float* __restrict__ sum,
                                               float* __restrict__ sumsq) {
  const int d = threadIdx.x;
  const size_t rowbase = (size_t)blockIdx.x * 64;
  float s = 0.0f, s2 = 0.0f;
  for (int i = 0; i < 64; ++i) {
    float x = V[(rowbase + i) * D_ + d];
    s += x; s2 += x * x;
  }
  atomicAdd(&sum[d], s);
  atomicAdd(&sumsq[d], s2);
}

__global__ __launch_bounds__(512) void k_bnfin(const float* __restrict__ sum,
                                               const float* __restrict__ sumsq,
                                               const float* __restrict__ gamma,
                                               const float* __restrict__ beta,
                                               float* __restrict__ scale,
                                               float* __restrict__ shift) {
  const int d = threadIdx.x;
  const float invN = 1.0f / (float)(B_ * T_);
  float mean = sum[d] * invN;
  float var  = sumsq[d] * invN - mean * mean;
  float sc   = gamma[d] * rsqrtf(var + 1e-5f);
  scale[d] = sc;
  shift[d] = beta[d] - mean * sc;
}

// ---------------------------------------------------------------------------
// Normalize V in place + vscore[b,d] = sum_t Vn[b,t,d]*w_v[t] + b_attn
// ---------------------------------------------------------------------------
__global__ __launch_bounds__(512) void k_norm(float* __restrict__ V,
                                              const float* __restrict__ scale,
                                              const float* __restrict__ shift,
                                              const float* __restrict__ wattn,
                                              const float* __restrict__ battn,
                                              float* __restrict__ vscore) {
  const int b = blockIdx.x, d = threadIdx.x;
  const float sc = scale[d], sh = shift[d];
  float acc = 0.0f;
  for (int t = 0; t < T_; ++t) {
    const size_t idx = ((size_t)b * T_ + t) * D_ + d;
    float x = V[idx] * sc + sh;
    V[idx] = x;
    acc += x * wattn[2 * H_ + t];
  }
  vscore[(size_t)b * D_ + d] = acc + battn[0];
}

// ---------------------------------------------------------------------------
// Per step: scores = vscore + (h.w_h + c.w_s); softmax over D; x_t = alpha*v_t
// One block per batch row; three LDS tree reductions.
// ---------------------------------------------------------------------------
__global__ __launch_bounds__(512) void k_attn(const float* __restrict__ Vn,
                                              const float* __restrict__ vscore,
                                              const float* __restrict__ h,
                                              const float* __restrict__ c,
                                              const float* __restrict__ wattn,
                                              float* __restrict__ xt, int t) {
  __shared__ float red[512];
  const int b = blockIdx.x, d = threadIdx.x;

  red[d] = h[(size_t)b * H_ + d] * wattn[d] + c[(size_t)b * H_ + d] * wattn[H_ + d];
  __syncthreads();
  for (int s = 256; s > 0; s >>= 1) { if (d < s) red[d] += red[d + s]; __syncthreads(); }
  const float sb = red[0];
  __syncthreads();

  const float score = vscore[(size_t)b * D_ + d] + sb;
  red[d] = score;
  __syncthreads();
  for (int s = 256; s > 0; s >>= 1) { if (d < s) red[d] = fmaxf(red[d], red[d + s]); __syncthreads(); }
  const float mx = red[0];
  __syncthreads();

  const float e = expf(score - mx);
  red[d] = e;
  __syncthreads();
  for (int s = 256; s > 0; s >>= 1) { if (d < s) red[d] += red[d + s]; __syncthreads(); }
  const float Z = red[0];

  xt[(size_t)b * D_ + d] = (e / Z) * Vn[((size_t)b * T_ + t) * D_ + d];
}

// ---------------------------------------------------------------------------
// gates[b,j] = x_t[b,:].W_ih[j,:] + h[b,:].W_hh[j,:] + (b_ih[j]+b_hh[j])
// M=256, N=2048, K=512(+512). 32x64 wave strip; one block covers all 256 rows.
// Grid: (1, 32). use_x==0 for the pad phase (h-only recurrence).
// ---------------------------------------------------------------------------
__global__ __launch_bounds__(256) void k_gates(const float* __restrict__ XT,
                                               const float* __restrict__ Hs,
                                               const float* __restrict__ Wih,
                                               const float* __restrict__ Whh,
                                               const float* __restrict__ bih,
                                               const float* __restrict__ bhh,
                                               float* __restrict__ gates,
                                               int use_x) {
  const int lane = threadIdx.x & 31;
  const int wave = threadIdx.x >> 5;
  const int lo = lane & 15;
  const int hi = lane >> 4;
  const int mb = blockIdx.x * 256 + wave * 32;
  const int nb = blockIdx.y * 64;
  const int koff = 2 * hi;

  v8f p0 = {}, p1 = {}, p2 = {}, p3 = {};
  v8f q0 = {}, q1 = {}, q2 = {}, q3 = {};

  if (use_x) {
    const float* a_row0 = XT  + (size_t)(mb +  0 + lo) * D_;
    const float* a_row1 = XT  + (size_t)(mb + 16 + lo) * D_;
    const float* b_row0 = Wih + (size_t)(nb +  0 + lo) * D_;
    const float* b_row1 = Wih + (size_t)(nb + 16 + lo) * D_;
    const float* b_row2 = Wih + (size_t)(nb + 32 + lo) * D_;
    const float* b_row3 = Wih + (size_t)(nb + 48 + lo) * D_;
    for (int kb = 0; kb < D_; kb += 4) {
      v2f a0 = *(const v2f*)(a_row0 + kb + koff);
      v2f a1 = *(const v2f*)(a_row1 + kb + koff);
      v2f b0 = *(const v2f*)(b_row0 + kb + koff);
      v2f b1 = *(const v2f*)(b_row1 + kb + koff);
      v2f b2 = *(const v2f*)(b_row2 + kb + koff);
      v2f b3 = *(const v2f*)(b_row3 + kb + koff);
      p0 = wmma4(a0, b0, p0);  q0 = wmma4(a1, b0, q0);
      p1 = wmma4(a0, b1, p1);  q1 = wmma4(a1, b1, q1);
      p2 = wmma4(a0, b2, p2);  q2 = wmma4(a1, b2, q2);
      p3 = wmma4(a0, b3, p3);  q3 = wmma4(a1, b3, q3);
    }
  }
  {
    const float* a_row0 = Hs  + (size_t)(mb +  0 + lo) * H_;
    const float* a_row1 = Hs  + (size_t)(mb + 16 + lo) * H_;
    const float* b_row0 = Whh + (size_t)(nb +  0 + lo) * H_;
    const float* b_row1 = Whh + (size_t)(nb + 16 + lo) * H_;
    const float* b_row2 = Whh + (size_t)(nb + 32 + lo) * H_;
    const float* b_row3 = Whh + (size_t)(nb + 48 + lo) * H_;
    for (int kb = 0; kb < H_; kb += 4) {
      v2f a0 = *(const v2f*)(a_row0 + kb + koff);
      v2f a1 = *(const v2f*)(a_row1 + kb + koff);
      v2f b0 = *(const v2f*)(b_row0 + kb + koff);
      v2f b1 = *(const v2f*)(b_row1 + kb + koff);
      v2f b2 = *(const v2f*)(b_row2 + kb + koff);
      v2f b3 = *(const v2f*)(b_row3 + kb + koff);
      p0 = wmma4(a0, b0, p0);  q0 = wmma4(a1, b0, q0);
      p1 = wmma4(a0, b1, p1);  q1 = wmma4(a1, b1, q1);
      p2 = wmma4(a0, b2, p2);  q2 = wmma4(a1, b2, q2);
      p3 = wmma4(a0, b3, p3);  q3 = wmma4(a1, b3, q3);
    }
  }

  const float bias0 = bih[nb +  0 + lo] + bhh[nb +  0 + lo];
  const float bias1 = bih[nb + 16 + lo] + bhh[nb + 16 + lo];
  const float bias2 = bih[nb + 32 + lo] + bhh[nb + 32 + lo];
  const float bias3 = bih[nb + 48 + lo] + bhh[nb + 48 + lo];
#pragma unroll
  for (int r = 0; r < 8; ++r) {
    const size_t m0 = (size_t)(mb + r + 8 * hi);
    const size_t m1 = m0 + 16;
    gates[m0 * G4 + nb +  0 + lo] = p0[r] + bias0;
    gates[m0 * G4 + nb + 16 + lo] = p1[r] + bias1;
    gates[m0 * G4 + nb + 32 + lo] = p2[r] + bias2;
    gates[m0 * G4 + nb + 48 + lo] = p3[r] + bias3;
    gates[m1 * G4 + nb +  0 + lo] = q0[r] + bias0;
    gates[m1 * G4 + nb + 16 + lo] = q1[r] + bias1;
    gates[m1 * G4 + nb + 32 + lo] = q2[r] + bias2;
    gates[m1 * G4 + nb + 48 + lo] = q3[r] + bias3;
  }
}

// ---------------------------------------------------------------------------
// LSTM pointwise cell + write h into output (B, T+cap, H)
// ---------------------------------------------------------------------------
__global__ __launch_bounds__(512) void k_cell(const float* __restrict__ gates,
                                              float* __restrict__ h,
                                              float* __restrict__ c,
                                              float* __restrict__ out,
                                              int t, int total_t) {
  const int b = blockIdx.x, d = threadIdx.x;
  const float* g = gates + (size_t)b * G4;
  const float ig = sigm(g[d]);
  const float fg = sigm(g[H_ + d]);
  const float gg = tanhf(g[2 * H_ + d]);
  const float og = sigm(g[3 * H_ + d]);
  const float cn = fg * c[(size_t)b * H_ + d] + ig * gg;
  const float hn = og * tanhf(cn);
  c[(size_t)b * H_ + d] = cn;
  h[(size_t)b * H_ + d] = hn;
  out[((size_t)b * total_t + t) * H_ + d] = hn;
}

// ---------------------------------------------------------------------------
// Host-side orchestration (all on `stream`, graph-capture safe)
// ---------------------------------------------------------------------------
extern "C" void kernel_launch(void* const* d_in, const int* in_sizes, int n_in,
                              void* d_out, int out_size, void* d_ws, size_t ws_size,
                              hipStream_t stream) {
  const float* video = (const float*)d_in[0];
  // d_in[1] = captions (int) — only its length matters
  const float* Wenc  = (const float*)d_in[2];
  const float* benc  = (const float*)d_in[3];
  const float* gamma = (const float*)d_in[4];
  const float* beta  = (const float*)d_in[5];
  const float* wattn = (const float*)d_in[6];
  const float* battn = (const float*)d_in[7];
  const float* Wih   = (const float*)d_in[8];
  const float* Whh   = (const float*)d_in[9];
  const float* bih   = (const float*)d_in[10];
  const float* bhh   = (const float*)d_in[11];
  float* out = (float*)d_out;

  const int caplen  = in_sizes[1] / B_;   // 20
  const int total_t = T_ + caplen;        // 80

  // Scratch layout (floats). ~35 MB total.
  float* ws = (float*)d_ws;
  size_t off = 0;
  float* V      = ws + off; off += (size_t)B_ * T_ * D_;   // 15360 x 512
  float* vscore = ws + off; off += (size_t)B_ * D_;
  float* xt     = ws + off; off += (size_t)B_ * D_;
  float* h      = ws + off; off += (size_t)B_ * H_;
  float* c      = ws + off; off += (size_t)B_ * H_;
  float* gates  = ws + off; off += (size_t)B_ * G4;
  float* sum    = ws + off; off += D_;
  float* sumsq  = ws + off; off += D_;
  float* scale  = ws + off; off += D_;
  float* shift  = ws + off; off += D_;
  (void)ws_size; (void)n_in; (void)out_size;

  k_zero<<<(B_ * H_ + 255) / 256, 256, 0, stream>>>(h, c, sum, sumsq);

  k_encode<<<dim3(B_ * T_ / 256, D_ / 64), 256, 0, stream>>>(video, Wenc, benc, V);
  k_stats<<<B_ * T_ / 64, 512, 0, stream>>>(V, sum, sumsq);
  k_bnfin<<<1, 512, 0, stream>>>(sum, sumsq, gamma, beta, scale, shift);
  k_norm<<<B_, 512, 0, stream>>>(V, scale, shift, wattn, battn, vscore);

  for (int t = 0; t < T_; ++t) {
    k_attn<<<B_, 512, 0, stream>>>(V, vscore, h, c, wattn, xt, t);
    k_gates<<<dim3(1, G4 / 64), 256, 0, stream>>>(xt, h, Wih, Whh, bih, bhh, gates, 1);
    k_cell<<<B_, 512, 0, stream>>>(gates, h, c, out, t, total_t);
  }
  for (int p = 0; p < caplen; ++p) {
    k_gates<<<dim3(1, G4 / 64), 256, 0, stream>>>(xt, h, Wih, Whh, bih, bhh, gates, 0);
    k_cell<<<B_, 512, 0, stream>>>(gates, h, c, out, T_ + p, total_t);
  }
}